// Attention_12489764897542
// MI455X (gfx1250) — compile-verified
//
#include <hip/hip_runtime.h>

typedef float v2f __attribute__((ext_vector_type(2)));
typedef float v8f __attribute__((ext_vector_type(8)));

#define S_LEN  2048
#define D_DIM  64
#define BH     16          // B*H
#define NEGINF (-1e9f)
#define SCALE  0.125f      // 1/sqrt(64)

#define KT_STRIDE 68       // LDS row stride (dwords), bank-conflict-free b64 frags
#define VT_STRIDE 72       // LDS row stride (dwords), disjoint banks for hi=0/1 b32 pairs
#define PT_STRIDE 68       // LDS row stride (dwords) for P tile (b64 frags)

// ---------------------------------------------------------------------------
// Kernel 1: scores = mask_fill(scale * Q @ K^T)
// grid = (256, 16): blockIdx.x = qblk*16 + kblk (128x128 region), blockIdx.y = bh
// block = 256 threads = 8 waves; K-tile staged once in LDS, shared by all waves.
// 4 independent WMMA accumulator chains per group.
// ---------------------------------------------------------------------------
__global__ __launch_bounds__(256) void qk_scores_kernel(
    const float* __restrict__ Q, const float* __restrict__ K,
    const int* __restrict__ mask, float* __restrict__ scores)
{
    __shared__ float ktile[128 * KT_STRIDE];   // 34.8 KB

    const int bh   = blockIdx.y;
    const int b    = bh >> 3;          // H = 8
    const int qblk = blockIdx.x >> 4;  // S/128 = 16
    const int kblk = blockIdx.x & 15;
    const int wave = threadIdx.x >> 5;
    const int lane = threadIdx.x & 31;
    const int m    = lane & 15;
    const int hi   = lane >> 4;

    const int q0 = qblk * 128 + wave * 16;
    const int k0 = kblk * 128;

    // Cooperative load of K tile (128 rows x 64 cols) into LDS, float4 chunks.
    {
        const float* kbase = K + ((size_t)bh * S_LEN + (size_t)k0) * D_DIM;
        const int tid = threadIdx.x;
#pragma unroll
        for (int r = 0; r < 8; ++r) {
            const int chunk = tid + 256 * r;      // 0..2047
            const int row   = chunk >> 4;         // 16 chunks per 64-float row
            const int col   = (chunk & 15) * 4;
            *(float4*)(&ktile[row * KT_STRIDE + col]) =
                *(const float4*)(kbase + (size_t)row * D_DIM + col);
        }
    }

    // Pre-load all 16 A fragments (16x64 Q slab, K-step = 4) from global.
    const float* qrow = Q + ((size_t)bh * S_LEN + (size_t)(q0 + m)) * D_DIM;
    v2f afrag[16];
#pragma unroll
    for (int j = 0; j < 16; ++j)
        afrag[j] = *(const v2f*)(qrow + 4 * j + 2 * hi);

    __syncthreads();

    for (int tg = 0; tg < 2; ++tg) {
        v8f acc[4] = {v8f{}, v8f{}, v8f{}, v8f{}};
#pragma unroll
        for (int j = 0; j < 16; ++j) {
#pragma unroll
            for (int t = 0; t < 4; ++t) {
                const int ldsrow = (tg * 4 + t) * 16 + m;   // B column == key row
                v2f bfrag = *(const v2f*)(&ktile[ldsrow * KT_STRIDE + 4 * j + 2 * hi]);
                acc[t] = __builtin_amdgcn_wmma_f32_16x16x4_f32(
                    false, afrag[j], false, bfrag, (short)0, acc[t], false, false);
            }
        }
        // scale + mask-fill + store (C/D layout: row = r + 8*hi, col = ncol)
#pragma unroll
        for (int t = 0; t < 4; ++t) {
            const int ncol = k0 + (tg * 4 + t) * 16 + m;
#pragma unroll
            for (int r = 0; r < 8; ++r) {
                const int qq = q0 + r + 8 * hi;
                const int mv = mask[((size_t)b * S_LEN + qq) * S_LEN + ncol];
                const float s = (mv == 0) ? NEGINF : acc[t][r] * SCALE;
                scores[((size_t)bh * S_LEN + qq) * S_LEN + ncol] = s;
            }
        }
    }
}

// ---------------------------------------------------------------------------
// Kernel 2: row softmax. grid = BH*S rows, block = 256 threads, 8 elems/thread
// ---------------------------------------------------------------------------
__global__ __launch_bounds__(256) void softmax_kernel(
    const float* __restrict__ scores, float* __restrict__ p)
{
    __shared__ float red[256];
    const size_t row = blockIdx.x;
    const float* src = scores + row * S_LEN;
    float*       dst = p + row * S_LEN;
    const int tid = threadIdx.x;

    float v[8];
    float lmax = -3.402823466e+38f;
#pragma unroll
    for (int i = 0; i < 8; ++i) {
        v[i] = src[tid + 256 * i];
        lmax = fmaxf(lmax, v[i]);
    }
    red[tid] = lmax; __syncthreads();
    for (int s = 128; s > 0; s >>= 1) {
        if (tid < s) red[tid] = fmaxf(red[tid], red[tid + s]);
        __syncthreads();
    }
    const float mrow = red[0];
    __syncthreads();

    float lsum = 0.f;
#pragma unroll
    for (int i = 0; i < 8; ++i) {
        v[i] = __expf(v[i] - mrow);
        lsum += v[i];
    }
    red[tid] = lsum; __syncthreads();
    for (int s = 128; s > 0; s >>= 1) {
        if (tid < s) red[tid] += red[tid + s];
        __syncthreads();
    }
    const float inv = 1.0f / red[0];
#pragma unroll
    for (int i = 0; i < 8; ++i)
        dst[tid + 256 * i] = v[i] * inv;
}

// ---------------------------------------------------------------------------
// Kernel 3: out = P @ V. grid = (S/64, BH), block = 128 threads = 4 waves.
// Both the 64x64 V chunk and the block's 64x64 P tile are staged in LDS each
// K-chunk: global side is fully-coalesced float4; fragment side is bank-
// conflict-free ds_load. Each wave: 16 q-rows x full D=64, 4 acc chains.
// ---------------------------------------------------------------------------
__global__ __launch_bounds__(128) void pv_kernel(
    const float* __restrict__ P, const float* __restrict__ V,
    float* __restrict__ out)
{
    __shared__ float vtile[64 * VT_STRIDE];    // 18.0 KB
    __shared__ float ptile[64 * PT_STRIDE];    // 17.4 KB

    const int bh   = blockIdx.y;
    const int wave = threadIdx.x >> 5;
    const int lane = threadIdx.x & 31;
    const int m    = lane & 15;
    const int hi   = lane >> 4;
    const int qb   = blockIdx.x * 64;          // block's first q-row
    const int q0   = qb + wave * 16;

    const float* pbase = P + ((size_t)bh * S_LEN + (size_t)qb) * S_LEN;
    const float* vmat  = V + (size_t)bh * S_LEN * D_DIM;

    v8f acc[4] = {v8f{}, v8f{}, v8f{}, v8f{}};

    for (int kb0 = 0; kb0 < S_LEN; kb0 += 64) {
        __syncthreads();
        // Stage V[kb0..kb0+63][0..63] and P[qb..qb+63][kb0..kb0+63] into LDS.
        const int tid = threadIdx.x;
#pragma unroll
        for (int r = 0; r < 8; ++r) {
            const int chunk = tid + 128 * r;      // 0..1023
            const int row   = chunk >> 4;
            const int col   = (chunk & 15) * 4;
            *(float4*)(&vtile[row * VT_STRIDE + col]) =
                *(const float4*)(vmat + (size_t)(kb0 + row) * D_DIM + col);
            *(float4*)(&ptile[row * PT_STRIDE + col]) =
                *(const float4*)(pbase + (size_t)row * S_LEN + kb0 + col);
        }
        __syncthreads();

#pragma unroll 4
        for (int kk = 0; kk < 64; kk += 4) {
            v2f a = *(const v2f*)(&ptile[(wave * 16 + m) * PT_STRIDE + kk + 2 * hi]);
            const float* vr = &vtile[(kk + 2 * hi) * VT_STRIDE + m];
#pragma unroll
            for (int t = 0; t < 4; ++t) {
                v2f bfrag;
                bfrag.x = vr[t * 16];               // V[kk+2*hi    ][16t + m]
                bfrag.y = vr[t * 16 + VT_STRIDE];   // V[kk+2*hi + 1][16t + m]
                acc[t] = __builtin_amdgcn_wmma_f32_16x16x4_f32(
                    false, a, false, bfrag, (short)0, acc[t], false, false);
            }
        }
    }

#pragma unroll
    for (int t = 0; t < 4; ++t)
#pragma unroll
        for (int r = 0; r < 8; ++r)
            out[((size_t)bh * S_LEN + (size_t)(q0 + r + 8 * hi)) * D_DIM + t * 16 + m] = acc[t][r];
}

// ---------------------------------------------------------------------------
extern "C" void kernel_launch(void* const* d_in, const int* in_sizes, int n_in,
                              void* d_out, int out_size, void* d_ws, size_t ws_size,
                              hipStream_t stream)
{
    (void)in_sizes; (void)n_in; (void)out_size; (void)d_ws; (void)ws_size;
    const float* Q    = (const float*)d_in[0];
    const float* K    = (const float*)d_in[1];
    const float* V    = (const float*)d_in[2];
    const int*   mask = (const int*)d_in[3];

    float* out    = (float*)d_out;
    float* p_attn = out + (size_t)BH * S_LEN * D_DIM;         // +2,097,152
    float* scores = p_attn + (size_t)BH * S_LEN * S_LEN;      // +67,108,864

    dim3 g1(256, BH);
    qk_scores_kernel<<<g1, 256, 0, stream>>>(Q, K, mask, scores);

    softmax_kernel<<<BH * S_LEN, 256, 0, stream>>>(scores, p_attn);

    dim3 g3(S_LEN / 64, BH);
    pv_kernel<<<g3, 128, 0, stream>>>(p_attn, V, out);
}